// GraphConvLayer_7404523619174
// MI455X (gfx1250) — compile-verified
//
#include <hip/hip_runtime.h>
#include <hip/hip_bf16.h>

typedef __attribute__((ext_vector_type(2))) float v2f;
typedef __attribute__((ext_vector_type(8))) float v8f;

#define WAVES_PER_BLOCK 8
#define LDA 68   // padded row pitch (floats) -> conflict-free ds_load_b64 frags

// ---------------------------------------------------------------------------
// Kernel 1: zero the aggregation buffer (lives in d_out)
// ---------------------------------------------------------------------------
__global__ void zero_kernel(float4* __restrict__ p, int n4) {
    int i = blockIdx.x * blockDim.x + threadIdx.x;
    if (i < n4) p[i] = make_float4(0.f, 0.f, 0.f, 0.f);
}

// ---------------------------------------------------------------------------
// Kernel 2: edge scatter  agg[tgt] += gw * x[src]
// 16 lanes per edge, float4 per lane -> one coalesced 256B row read per edge,
// 4 global_atomic_add_f32 per lane. x and agg both fit in the 192MB L2.
// ---------------------------------------------------------------------------
__global__ void edge_scatter_kernel(const float* __restrict__ x,
                                    const long long* __restrict__ edge_index,
                                    const float* __restrict__ edge_attr,
                                    const float* __restrict__ gaussian_param,
                                    float* __restrict__ agg,
                                    int n_edges) {
    int t = blockIdx.x * blockDim.x + threadIdx.x;
    int e  = t >> 4;        // edge id
    int c4 = t & 15;        // float4 chunk within the 64-ch row
    if (e >= n_edges) return;

    long long s = edge_index[e];            // src row of (2, E) int64
    long long d = edge_index[n_edges + e];  // tgt row

    float dist = edge_attr[e * 3];          // edge_attr[:,0]
    float g    = gaussian_param[0];
    float gw   = __expf(-(dist * dist) / (g * g + 1e-8f));

    float4 v = ((const float4*)(x + s * 64))[c4];
    float* o = agg + d * 64 + (c4 << 2);
    unsafeAtomicAdd(o + 0, v.x * gw);
    unsafeAtomicAdd(o + 1, v.y * gw);
    unsafeAtomicAdd(o + 2, v.z * gw);
    unsafeAtomicAdd(o + 3, v.w * gw);
}

// ---------------------------------------------------------------------------
// Kernel 3: out = (agg + x) @ W + bias, in place (agg == out buffer).
// 8 waves/block, 16 rows/wave -> 128 rows/block. Each wave:
//   A tile (16x64, fp32) in LDS, W^T (64x64) in LDS shared by block,
//   K-loop of 16 steps of V_WMMA_F32_16X16X4_F32 over 4 N-tiles.
// Each block reads/writes only its own rows; kernel boundary syncs phases.
// ---------------------------------------------------------------------------
__global__ void node_gemm_kernel(const float* __restrict__ x,
                                 const float* __restrict__ W,
                                 const float* __restrict__ bias,
                                 float* __restrict__ out,
                                 int n_nodes) {
    __shared__ float sA[WAVES_PER_BLOCK * 16 * LDA]; // per-wave 16x64 A tiles
    __shared__ float sW[64 * LDA];                   // W transposed: Wt[n][k]
    __shared__ float sBias[64];

    const int tid  = threadIdx.x;
    const int wave = tid >> 5;
    const int lane = tid & 31;
    const int half = lane >> 4;   // 0: lanes 0-15, 1: lanes 16-31
    const int l    = lane & 15;

    const int wave_row = blockIdx.x * (WAVES_PER_BLOCK * 16) + wave * 16;

    // --- stage W^T into LDS (coalesced global read, once per block) ---
    for (int i = tid; i < 64 * 64; i += 256) {
        int k = i >> 6, n = i & 63;
        sW[n * LDA + k] = W[i];
    }
    if (tid < 64) sBias[tid] = bias[tid];

    // --- stage A = agg(out) + x for this wave's 16 rows ---
    float* myA = &sA[wave * 16 * LDA];
    for (int i = lane; i < 16 * 16; i += 32) {      // 16 rows x 16 float4
        int r = i >> 4, c4 = i & 15;
        int row = wave_row + r;
        int rc  = row < n_nodes ? row : n_nodes - 1; // clamp stays in-block
        float4 a = ((const float4*)(out + (long long)rc * 64))[c4];
        float4 b = ((const float4*)(x   + (long long)rc * 64))[c4];
        float* p = &myA[r * LDA + (c4 << 2)];
        p[0] = a.x + b.x; p[1] = a.y + b.y; p[2] = a.z + b.z; p[3] = a.w + b.w;
    }
    __syncthreads();

    // --- WMMA: D(16x16) += A(16x4) x B(4x16), 16 K-steps, 4 N-tiles ---
    v8f acc0 = {}, acc1 = {}, acc2 = {}, acc3 = {};
#pragma unroll
    for (int k = 0; k < 16; ++k) {
        // 32-bit A 16x4 layout: lane l holds row M=l; VGPR0/1 = K {0,1} for
        // lanes 0-15, K {2,3} for lanes 16-31.
        v2f a = *(const v2f*)&myA[l * LDA + 4 * k + 2 * half];
        // B 4x16 from W^T: lane half selects K pair, lane selects column N.
        v2f b0 = *(const v2f*)&sW[(0  + l) * LDA + 4 * k + 2 * half];
        v2f b1 = *(const v2f*)&sW[(16 + l) * LDA + 4 * k + 2 * half];
        v2f b2 = *(const v2f*)&sW[(32 + l) * LDA + 4 * k + 2 * half];
        v2f b3 = *(const v2f*)&sW[(48 + l) * LDA + 4 * k + 2 * half];
        acc0 = __builtin_amdgcn_wmma_f32_16x16x4_f32(false, a, false, b0, (short)0, acc0, false, false);
        acc1 = __builtin_amdgcn_wmma_f32_16x16x4_f32(false, a, false, b1, (short)0, acc1, false, false);
        acc2 = __builtin_amdgcn_wmma_f32_16x16x4_f32(false, a, false, b2, (short)0, acc2, false, false);
        acc3 = __builtin_amdgcn_wmma_f32_16x16x4_f32(false, a, false, b3, (short)0, acc3, false, false);
    }

    // --- store: C/D layout VGPR p: lanes0-15 -> M=p, lanes16-31 -> M=8+p ---
    const int row_base = wave_row + 8 * half;
#pragma unroll
    for (int p = 0; p < 8; ++p) {
        int row = row_base + p;
        if (row < n_nodes) {
            long long ro = (long long)row * 64;
            out[ro +  0 + l] = acc0[p] + sBias[ 0 + l];
            out[ro + 16 + l] = acc1[p] + sBias[16 + l];
            out[ro + 32 + l] = acc2[p] + sBias[32 + l];
            out[ro + 48 + l] = acc3[p] + sBias[48 + l];
        }
    }
}

// ---------------------------------------------------------------------------
extern "C" void kernel_launch(void* const* d_in, const int* in_sizes, int n_in,
                              void* d_out, int out_size, void* d_ws, size_t ws_size,
                              hipStream_t stream) {
    const float*     x   = (const float*)d_in[0];
    const long long* ei  = (const long long*)d_in[1];   // int64 (2, E)
    const float*     ea  = (const float*)d_in[2];       // (E, 3)
    const float*     W   = (const float*)d_in[3];       // (64, 64)
    const float*     bs  = (const float*)d_in[4];       // (64,)
    const float*     gp  = (const float*)d_in[5];       // (1,)
    float*           out = (float*)d_out;

    const int n_nodes = in_sizes[0] / 64;
    const int n_edges = in_sizes[2] / 3;

    // Phase 0: zero agg (== d_out)
    int n4 = (n_nodes * 64) / 4;
    zero_kernel<<<(n4 + 255) / 256, 256, 0, stream>>>((float4*)out, n4);

    // Phase 1: edge scatter with fp32 atomics (16 lanes per edge)
    long long threads = (long long)n_edges * 16;
    int blocks = (int)((threads + 255) / 256);
    edge_scatter_kernel<<<blocks, 256, 0, stream>>>(x, ei, ea, gp, out, n_edges);

    // Phase 2: out = (agg + x) @ W + bias via V_WMMA_F32_16X16X4_F32
    int gblocks = (n_nodes + WAVES_PER_BLOCK * 16 - 1) / (WAVES_PER_BLOCK * 16);
    node_gemm_kernel<<<gblocks, 256, 0, stream>>>(x, W, bs, out, n_nodes);
}